// CURVGT_65833258713572
// MI455X (gfx1250) — compile-verified
//
#include <hip/hip_runtime.h>
#include <hip/hip_bf16.h>
#include <stdint.h>

// Mesh constants (match reference)
#define V_  150000
#define E_  900000
#define B_  2
#define N_  (B_ * V_)     // 300000
#define BE_ (B_ * E_)     // 1800000

typedef __attribute__((ext_vector_type(4))) unsigned int u32x4;
typedef __attribute__((ext_vector_type(8))) int          i32x8;
typedef __attribute__((ext_vector_type(4))) int          i32x4;

// ---- monotone float <-> u32 mapping for atomic segment-max ----
__device__ __forceinline__ unsigned fmap(float f) {
    unsigned u = __float_as_uint(f);
    return (u & 0x80000000u) ? ~u : (u | 0x80000000u);
}
__device__ __forceinline__ float funmap(unsigned m) {
    unsigned u = (m & 0x80000000u) ? (m & 0x7fffffffu) : ~m;
    return __uint_as_float(u);
}

// ---- Tensor Data Mover: 2-D row-tile load (global -> LDS), clang-23 6-arg form ----
// D# group0: [1:0]=count=1, [63:32]=lds_addr, [120:64]=global_addr, [127:126]=type=2
// D# group1: [17:16]=data_size(2 => 4B), [79:48]=tensor_dim0, [111:80]=tensor_dim1,
//            [127:112]=tile_dim0, [143:128]=tile_dim1, [207:160]=tensor_dim0_stride
__device__ __forceinline__ void tdm_load_rows(uint32_t lds_off, const void* gptr,
                                              uint32_t row_dwords, uint32_t rows,
                                              uint32_t tensor_rows) {
    uint64_t ga = (uint64_t)(uintptr_t)gptr;
    u32x4 g0;
    g0.x = 1u;                                            // count=1, user descriptor
    g0.y = lds_off;                                       // LDS byte address
    g0.z = (uint32_t)ga;                                  // global_addr[31:0]
    g0.w = ((uint32_t)(ga >> 32) & 0x01FFFFFFu) | 0x80000000u; // addr[56:32] | type=2
    i32x8 g1;
    g1.s0 = (int)(2u << 16);                              // data_size = 4 bytes
    g1.s1 = (int)((row_dwords & 0xFFFFu) << 16);          // tensor_dim0[15:0]
    g1.s2 = (int)(((row_dwords >> 16) & 0xFFFFu) |        // tensor_dim0[31:16]
                  ((tensor_rows & 0xFFFFu) << 16));       // tensor_dim1[15:0]
    g1.s3 = (int)(((tensor_rows >> 16) & 0xFFFFu) |       // tensor_dim1[31:16]
                  ((row_dwords & 0xFFFFu) << 16));        // tile_dim0 = full row
    g1.s4 = (int)(rows & 0xFFFFu);                        // tile_dim1, tile_dim2=0
    g1.s5 = (int)row_dwords;                              // tensor_dim0_stride[31:0]
    g1.s6 = 0;                                            // stride hi / dim1_stride
    g1.s7 = 0;
    i32x4 z4 = {0, 0, 0, 0};
    i32x8 z8 = {0, 0, 0, 0, 0, 0, 0, 0};
    __builtin_amdgcn_tensor_load_to_lds(g0, g1, z4, z4, z8, 0);
}

// ---- Pass 0: zero smax-bits / denom / accum (5N dwords) ----
__global__ void CURVGT_init_kernel(unsigned* __restrict__ ws) {
    int i = blockIdx.x * 256 + threadIdx.x;
    if (i < 5 * N_) ws[i] = 0u;   // 0u == mapped(-inf) for smax, 0.0f for floats
}

// ---- Pass 1: per-edge score + pre-message; atomic segment max ----
// Emits one packed b128 record {pre_msg.xyz, score} per edge.
__global__ __launch_bounds__(256) void CURVGT_edge_score_kernel(
    const float* __restrict__ x, const int* __restrict__ ei,
    const float* __restrict__ edge_attrs, const float* __restrict__ h2f,
    const float* __restrict__ hpt, const int* __restrict__ omask,
    const int* __restrict__ bmap, const float* __restrict__ kK,
    const float* __restrict__ kK2, const float* __restrict__ ap,
    const float* __restrict__ att, const float* __restrict__ W1,
    const float* __restrict__ b1, const float* __restrict__ c,
    unsigned* __restrict__ smax_bits, float4* __restrict__ rec) {
    __shared__ float s_ea[256 * 20];
    __shared__ float s_h2[256 * 6];
    __shared__ float s_hp[256 * 4];

    const int e0   = blockIdx.x * 256;
    const int rows = (E_ - e0) < 256 ? (E_ - e0) : 256;

    if (threadIdx.x == 0) {   // one wave reaches this via execz branch; TDM ignores EXEC
        tdm_load_rows((uint32_t)(uintptr_t)&s_ea[0], edge_attrs + (size_t)e0 * 20, 20, rows, E_);
        tdm_load_rows((uint32_t)(uintptr_t)&s_h2[0], h2f        + (size_t)e0 * 6,   6, rows, E_);
        tdm_load_rows((uint32_t)(uintptr_t)&s_hp[0], hpt        + (size_t)e0 * 4,   4, rows, E_);
        __builtin_amdgcn_s_wait_tensorcnt(0);
    }
    __syncthreads();

    const int t = threadIdx.x;
    const int e = e0 + t;
    if (e >= E_) return;
    const int ee  = blockIdx.y * E_ + e;          // tiled edge id, < 1.8M
    const int src = ei[ee];
    const int dst = ei[BE_ + ee];

    const float* er = &s_ea[t * 20];
    const float theta = er[9];
    const float cth = __cosf(theta), sth = __sinf(theta);

    float xj0 = x[src * 3 + 0], xj1 = x[src * 3 + 1], xj2 = x[src * 3 + 2];
    float xi0 = x[dst * 3 + 0], xi1 = x[dst * 3 + 1], xi2 = x[dst * 3 + 2];

    // spherical transport
    const float a  = er[11] * xj0 + er[12] * xj1 + er[13] * xj2;
    const float b  = er[14] * xj0 + er[15] * xj1 + er[16] * xj2;
    const float ac = a * cth, as = a * sth;
    float p0, p1, p2;
    const int m = omask[e];
    if (m == 1) {
        p0 = ac * er[11] + as * er[17] + b * er[14];
        p1 = ac * er[12] + as * er[18] + b * er[15];
        p2 = ac * er[13] + as * er[19] + b * er[16];
    } else if (m == -1) {
        const float* fr = &s_h2[t * 6];
        const float* Tr = &s_hp[t * 4];
        const float a2 = fr[0] * xj0 + fr[1] * xj1 + fr[2] * xj2;
        const float b2 = fr[3] * xj0 + fr[4] * xj1 + fr[5] * xj2;
        const float l0 = Tr[0] * a2 + Tr[1] * b2;
        const float l1 = Tr[2] * a2 + Tr[3] * b2;
        p0 = fr[0] * l0 + fr[3] * l1;
        p1 = fr[1] * l0 + fr[4] * l1;
        p2 = fr[2] * l0 + fr[5] * l1;
    } else {
        p0 = xj0; p1 = xj1; p2 = xj2;
    }

    // tree-broadcast curvature gather (L2-resident)
    int dv = dst; if (dv >= V_) dv -= V_;
    const int r = bmap[dv];
    const float* kr  = kK  + (size_t)r * 9;
    const float* kr2 = kK2 + (size_t)r * 9;
    const float* ar  = ap  + (size_t)r * 6;
    const float m10 = kr[0]*p0 + kr[1]*p1 + kr[2]*p2;
    const float m11 = kr[3]*p0 + kr[4]*p1 + kr[5]*p2;
    const float m12 = kr[6]*p0 + kr[7]*p1 + kr[8]*p2;
    const float m20 = kr2[0]*p0 + kr2[1]*p1 + kr2[2]*p2;
    const float m21 = kr2[3]*p0 + kr2[4]*p1 + kr2[5]*p2;
    const float m22 = kr2[6]*p0 + kr2[7]*p1 + kr2[8]*p2;

    // layernormA over the 6 values {m1, m2}
    const float mu = (m10 + m11 + m12 + m20 + m21 + m22) * (1.0f / 6.0f);
    float d0=m10-mu, d1=m11-mu, d2=m12-mu, d3=m20-mu, d4=m21-mu, d5=m22-mu;
    const float var = (d0*d0 + d1*d1 + d2*d2 + d3*d3 + d4*d4 + d5*d5) * (1.0f / 6.0f);
    const float rs = rsqrtf(var + 1e-5f);
    // s = einsum('cd,dc') with attn rows, then layernormB over 2 values
    const float s0 = (ar[0]*d0 + ar[1]*d1 + ar[2]*d2) * rs;
    const float s1 = (ar[3]*d3 + ar[4]*d4 + ar[5]*d5) * rs;
    const float mu2 = 0.5f * (s0 + s1);
    const float dd  = s0 - mu2;
    const float rs2 = rsqrtf(dd * dd + 1e-5f);      // var over 2 = dd^2
    const float s0n = dd * rs2, s1n = -dd * rs2;

    const float lin = W1[0] * s0n + W1[1] * s1n + b1[0];
    const float g = att[0]*xi0 + att[1]*xi1 + att[2]*xi2
                  + att[3]*p0  + att[4]*p1  + att[5]*p2;
    const float gat = (g > 0.0f) ? g : 0.2f * g;
    const float score = gat + lin;

    const float c0 = c[0], c1 = c[1];
    float4 rv;
    rv.x = p0 + c0 * m10 + c1 * m20;
    rv.y = p1 + c0 * m11 + c1 * m21;
    rv.z = p2 + c0 * m12 + c1 * m22;
    rv.w = score;
    rec[ee] = rv;                                   // one global_store_b128 per edge
    atomicMax(&smax_bits[dst], fmap(score));
}

// ---- Pass 2: exp-weights, accumulate denom and weighted messages ----
__global__ void CURVGT_edge_accum_kernel(
    const int* __restrict__ ei, const float4* __restrict__ rec,
    const unsigned* __restrict__ smax_bits,
    float* __restrict__ denom, float* __restrict__ accum) {
    int ee = blockIdx.x * 256 + threadIdx.x;
    if (ee >= BE_) return;
    const int dst = ei[BE_ + ee];
    const float4 r = rec[ee];                       // one global_load_b128 per edge
    const float w = __expf(r.w - funmap(smax_bits[dst]));
    atomicAdd(&denom[dst], w);
    atomicAdd(&accum[dst * 3 + 0], w * r.x);
    atomicAdd(&accum[dst * 3 + 1], w * r.y);
    atomicAdd(&accum[dst * 3 + 2], w * r.z);
}

// ---- Pass 3: normalize per node ----
__global__ void CURVGT_node_kernel(const float* __restrict__ denom,
                                   const float* __restrict__ accum,
                                   float* __restrict__ out) {
    int n = blockIdx.x * 256 + threadIdx.x;
    if (n >= N_) return;
    const float inv = 1.0f / (denom[n] + 1e-16f);
    out[n * 3 + 0] = accum[n * 3 + 0] * inv;
    out[n * 3 + 1] = accum[n * 3 + 1] * inv;
    out[n * 3 + 2] = accum[n * 3 + 2] * inv;
}

extern "C" void kernel_launch(void* const* d_in, const int* in_sizes, int n_in,
                              void* d_out, int out_size, void* d_ws, size_t ws_size,
                              hipStream_t stream) {
    const float* x    = (const float*)d_in[0];
    const int*   ei   = (const int*)  d_in[1];
    const float* ea   = (const float*)d_in[2];
    const float* h2f  = (const float*)d_in[3];
    const float* hpt  = (const float*)d_in[4];
    const int*   om   = (const int*)  d_in[5];
    const int*   bmap = (const int*)  d_in[6];
    const float* kK   = (const float*)d_in[7];
    const float* kK2  = (const float*)d_in[8];
    const float* ap   = (const float*)d_in[9];
    const float* att  = (const float*)d_in[10];
    const float* W1   = (const float*)d_in[11];
    const float* b1   = (const float*)d_in[12];
    const float* c    = (const float*)d_in[13];
    float* out = (float*)d_out;

    // workspace layout (dwords): [smax_bits N][denom N][accum 3N][rec 4*BE]
    float*    wsf       = (float*)d_ws;
    unsigned* smax_bits = (unsigned*)wsf;
    float*    denom     = wsf + N_;
    float*    accum     = wsf + 2 * N_;
    float4*   rec       = (float4*)(wsf + 5 * (size_t)N_);   // 6MB offset, 16B aligned

    CURVGT_init_kernel<<<(5 * N_ + 255) / 256, 256, 0, stream>>>(smax_bits);

    dim3 g1((E_ + 255) / 256, B_);
    CURVGT_edge_score_kernel<<<g1, 256, 0, stream>>>(
        x, ei, ea, h2f, hpt, om, bmap, kK, kK2, ap, att, W1, b1, c,
        smax_bits, rec);

    CURVGT_edge_accum_kernel<<<(BE_ + 255) / 256, 256, 0, stream>>>(
        ei, rec, smax_bits, denom, accum);

    CURVGT_node_kernel<<<(N_ + 255) / 256, 256, 0, stream>>>(denom, accum, out);
}